// model_43138651521353
// MI455X (gfx1250) — compile-verified
//
#include <hip/hip_runtime.h>
#include <math.h>
#include <stdint.h>

// ---------------------------------------------------------------------------
// Stopped-diffusion Monte-Carlo estimator for MI455X (gfx1250, wave32).
//   B = 524288 samples, D = 2, N = 50 steps.
//   dw layout [B, D, N]  -> each sample's 100 floats are contiguous (400 B).
// Strategy: per-lane async staging of the 400-B dw slice into private LDS
// (global_load_async_to_lds_b128, CDNA5 async path, NT temporal hint),
// s_wait_asynccnt, then a rolled 50-step loop reading from LDS. The Ferrari
// quartic solve runs only on steps where ballot(mask) != 0; waves break out
// early once all 32 lanes have exited (state is frozen after exit => exact).
// ---------------------------------------------------------------------------

#define TPB    128
#define NSTEP  50

__device__ __forceinline__ float ferrari_exit_frac(float x, float y,
                                                   float drx, float dry,
                                                   float dfx, float dfy,
                                                   bool mask) {
    // quartic coefficients a t^4 + b t^3 + c t^2 + d t + e, t = sqrt(rho)
    float a  = drx * drx + dry * dry;
    float bq = 2.0f * (drx * dfx + dry * dfy);
    float c  = 2.0f * (drx * x + dry * y) + dfx * dfx + dfy * dfy;
    float d  = 2.0f * (dfx * x + dfy * y);
    float e  = x * x + y * y - 1.0f;           // R2 == 1
    // dummy (generic) quartic on non-exit lanes keeps all divisions finite
    a  = mask ? a  : 1.0f;
    bq = mask ? bq : 2.0f;
    c  = mask ? c  : 3.0f;
    d  = mask ? d  : 4.0f;
    e  = mask ? e  : 5.0f;

    float a2 = a * a;
    float a3 = a2 * a;
    float p  = (8.0f * a * c - 3.0f * bq * bq) / (8.0f * a2);
    float q  = (bq * bq * bq - 4.0f * a * bq * c + 8.0f * a2 * d) / (8.0f * a3);
    float sign_q = (q > 0.0f) ? 1.0f : ((q < 0.0f) ? -1.0f : 0.0f);

    float D0 = c * c - 3.0f * bq * d + 12.0f * a * e;
    float D1 = 2.0f * c * c * c - 9.0f * bq * c * d + 27.0f * bq * bq * e
             + 27.0f * a * d * d - 72.0f * a * c * e;
    float D2 = D1 * D1 - 4.0f * D0 * D0 * D0;
    bool  sel = (D2 >= 0.0f);                   // ceil((sign(D2)+1)/2) == 1

    float QQ = 0.5f * (D1 + sqrtf(fabsf(D2)));
    float Q  = cbrtf(QQ);                       // sign(QQ)*|QQ|^(1/3)
    float Sp = 0.5f * sqrtf(fabsf((Q + D0 / Q) / (3.0f * a)
                                  - (2.0f / 3.0f) * p));
    float ad0  = fabsf(D0);
    float sad0 = sqrtf(ad0);
    float phi  = acosf(fminf(fmaxf(D1 * 0.5f / (ad0 * sad0), -1.0f), 1.0f));
    float Sm = 0.5f * sqrtf(fabsf(2.0f * sad0 * cosf(phi * (1.0f / 3.0f))
                                  / (3.0f * a) - (2.0f / 3.0f) * p));
    float S  = sel ? Sp : Sm;

    float aqS = fabsf(q / S);
    float b4a = bq / (4.0f * a);
    float t0  = -4.0f * S * S - 2.0f * p + aqS;
    float sr  = 0.5f * sqrtf(fabsf(t0)) - b4a - sign_q * S;
    float t1  = -4.0f * S * S - 2.0f * p - aqS;
    float nsr = 0.5f * sqrtf(fabsf(t1)) - b4a + sign_q * S;
    sr = ((1.0f - sr) * sr < 0.0f) ? nsr : sr;  // root sanity check
    return sr * sr;
}

__global__ __launch_bounds__(TPB) void sde_exit_kernel(
        const float* __restrict__ x0,
        const float* __restrict__ dw,
        const float* __restrict__ u,
        float* __restrict__ out,
        int Bn) {
    // 128 threads * 100 floats = 51200 B of LDS, thread-private slices
    __shared__ float smem[TPB * 2 * NSTEP];

    const int b = blockIdx.x * TPB + threadIdx.x;

    // ---- CDNA5 async staging: 25 x b128 copies of this lane's dw slice ----
    unsigned lbase = (unsigned)(uintptr_t)(&smem[0])
                   + (unsigned)(threadIdx.x * (2 * NSTEP * 4));
    unsigned long long ga =
        (unsigned long long)(uintptr_t)(dw + (size_t)b * (2 * NSTEP));
#pragma unroll
    for (int k = 0; k < (2 * NSTEP * 4) / 16; ++k) {
        // INST_OFFSET is added to BOTH the LDS dest and the global address
        asm volatile(
            "global_load_async_to_lds_b128 %0, %1, off offset:%2 th:TH_LOAD_NT"
            :: "v"(lbase), "v"(ga), "i"(k * 16) : "memory");
    }

    // overlap: scalar/uniform + per-lane state loads while the DMA flies
    const float uu   = u[0];
    const float u2p2 = uu * uu + 2.0f;
    float xv = x0[2 * b + 0];
    float yv = x0[2 * b + 1];

    constexpr float DT    = 0.2f / (float)NSTEP;   // 0.004
    constexpr float SIGMA = 1.41421356237309515f;  // sqrt(2)
    constexpr float GDT   = DT;                    // gamma * dt, gamma == 1

    float flag = 1.0f;
    float csum = 0.0f;
    float acc  = 0.0f;

    asm volatile("s_wait_asynccnt 0x0" ::: "memory");

    const float* tl = &smem[threadIdx.x * 2 * NSTEP];

    for (int t = 0; t < NSTEP; ++t) {
        float dwx = tl[t];              // ds_load, 2-way bank conflict max
        float dwy = tl[NSTEP + t];

        float drx = uu * xv * DT;
        float dry = uu * yv * DT;
        float dfx = SIGMA * dwx;
        float dfy = SIGMA * dwy;

        float xt = xv + drx + dfx;
        float yt = yv + dry + dfy;
        float Exit = (xt * xt + yt * yt - 1.0f >= 0.0f) ? 1.0f : 0.0f;
        bool  mask = (flag * Exit) == 1.0f;     // exits this step, still alive

        float rho = 0.0f;
        if (__builtin_amdgcn_ballot_w32(mask))  // heavy path: wave-uniform skip
            rho = ferrari_exit_frac(xv, yv, drx, dry, dfx, dfy, mask);

        float coef = mask ? rho : (1.0f - Exit) * flag;

        // discounted running-cost term uses |x|^2 BEFORE the step
        float sq = xv * xv + yv * yv;
        acc += coef * (u2p2 * sq - 4.0f) * DT * expf(-GDT * csum);
        csum += coef;

        float sc = (coef > 0.0f) ? sqrtf(coef) : 0.0f;  // safe_sqrt
        xv += drx * coef + dfx * sc;
        yv += dry * coef + dfy * sc;
        flag *= (1.0f - Exit);

        // all lanes exited -> state frozen for remaining steps (exact)
        if (__builtin_amdgcn_ballot_w32(flag != 0.0f) == 0u) break;
    }

    acc += expf(-GDT * csum) * (xv * xv + yv * yv);  // terminal payoff
    if (b < Bn) out[b] = acc;
}

extern "C" void kernel_launch(void* const* d_in, const int* in_sizes, int n_in,
                              void* d_out, int out_size, void* d_ws, size_t ws_size,
                              hipStream_t stream) {
    (void)n_in; (void)d_ws; (void)ws_size; (void)out_size;
    const float* x0 = (const float*)d_in[0];   // [B, 2]
    const float* dw = (const float*)d_in[1];   // [B, 2, 50]
    const float* u  = (const float*)d_in[2];   // [1]
    float* out = (float*)d_out;                // [B, 1]

    const int Bn = in_sizes[0] / 2;            // 524288
    dim3 grid((Bn + TPB - 1) / TPB), block(TPB);
    sde_exit_kernel<<<grid, block, 0, stream>>>(x0, dw, u, out, Bn);
}